// ConsensusAttention_10342281248995
// MI455X (gfx1250) — compile-verified
//
#include <hip/hip_runtime.h>
#include <math.h>

// ConsensusAttention for MI455X (gfx1250), wave32 + fp32 WMMA.
// levels: [8, 1024, 6, 128] fp32 ; out: same shape.
// Local attention on a 32x32 token grid, radius 2 (dy^2+dx^2 <= 4), self logit = -5e-4.
//
// Softmax stabilization uses the analytic per-row bound  max_j sim[M,j] <= ||q_M||/sqrt(d)
// (keys are unit-normalized), so no running max / rescaling / cross-lane reductions are
// needed; row sums come out of an extra WMMA against an all-ones B operand.

typedef __attribute__((ext_vector_type(2))) float v2f;
typedef __attribute__((ext_vector_type(8))) float v8f;

#define BATCH    8
#define NTOK     1024
#define NLVL     6
#define DIM      128
#define GRID_W   32
#define SCALE    0.08838834764831845f       /* 1/sqrt(128) */
#define MAXNEG   -3.4028234663852886e38f    /* -FLT_MAX */
#define SELF_VAL -0.0005f

__global__ __launch_bounds__(32)
void consensus_attn_kernel(const float* __restrict__ levels, float* __restrict__ out) {
    __shared__ float pbuf[16 * 16];   // P tile staging: C-layout -> A-layout transpose
    __shared__ float bounds[16];      // per-query-row softmax shift

    const int lane = threadIdx.x;     // 0..31
    const int half = lane >> 4;       // 0 or 1
    const int ln16 = lane & 15;

    const int tile = blockIdx.x;      // 0..63 : 16-query tile within the (b,l) plane
    const int qy   = tile >> 1;
    const int qx0  = (tile & 1) << 4;
    const int l    = blockIdx.y;
    const int b    = blockIdx.z;

    // ---- Preload Q A-frags (A 16x4 per chunk: lane -> M = ln16; VGPR0 K=half*2, VGPR1 K+1)
    //      and accumulate ||q_M||^2 from the same registers.
    const int    qtok = qy * GRID_W + qx0 + ln16;
    const float* qrow = levels + ((size_t)(b * NTOK + qtok) * NLVL + l) * DIM;
    v2f   aq[32];
    float qn = 0.0f;
#pragma unroll
    for (int kc = 0; kc < 32; ++kc) {
        aq[kc] = *(const v2f*)(qrow + kc * 4 + half * 2);
        qn += aq[kc].x * aq[kc].x + aq[kc].y * aq[kc].y;
    }
    qn += __shfl_xor(qn, 16);                       // full ||q_M||^2, M = ln16
    bounds[ln16] = sqrtf(qn) * SCALE;               // logit upper bound for row M

    // Move bound from A-layout (lane = M) to C-layout (M = r + half*8); wave-local
    // LDS is processed in order, so no barrier is needed in a single-wave block.
    float mrow[8];
#pragma unroll
    for (int r = 0; r < 8; ++r)
        mrow[r] = bounds[half * 8 + r];

    // ---- accumulators: 128 output dims (8 x 16x16 tiles) + row-sum tile
    v8f O[8];
#pragma unroll
    for (int dc = 0; dc < 8; ++dc) O[dc] = (v8f){};
    v8f Osum = (v8f){};

    const v2f onesb = {1.0f, 1.0f};

    for (int ky = qy - 2; ky <= qy + 2; ++ky) {
        if (ky < 0 || ky >= GRID_W) continue;
        const int dy = qy - ky;
        for (int kxt = 0; kxt < 2; ++kxt) {
            const int kx0  = kxt << 4;
            const int ktok = ky * GRID_W + kx0 + ln16;  // this lane's key column N = ln16
            const float* krow = levels + ((size_t)(b * NTOK + ktok) * NLVL + l) * DIM;

            // ---- sim tile = Q @ levels^T, plus per-key norm^2 from the same loads
            v8f   C    = (v8f){};
            float nsum = 0.0f;
#pragma unroll
            for (int kc = 0; kc < 32; ++kc) {
                v2f bv = *(const v2f*)(krow + kc * 4 + half * 2);   // B-frag: N=ln16, K pair
                nsum += bv.x * bv.x + bv.y * bv.y;
                C = __builtin_amdgcn_wmma_f32_16x16x4_f32(
                        false, aq[kc], false, bv, (short)0, C, false, false);
            }
            // lanes L and L^16 each hold half of key ln16's |v|^2
            float nrm2     = nsum + __shfl_xor(nsum, 16);
            float colscale = SCALE / fmaxf(sqrtf(nrm2), 1e-12f);

            // ---- scale, mask, exponentiate with the fixed per-row shift
            float p[8];
#pragma unroll
            for (int r = 0; r < 8; ++r) {
                float cv = C[r] * colscale;
                int M  = r + half * 8;
                int dx = (qx0 + M) - (kx0 + ln16);
                int d2 = dy * dy + dx * dx;
                if (d2 > 4)       cv = MAXNEG;     // exp -> 0
                else if (d2 == 0) cv = SELF_VAL;
                p[r] = __expf(cv - mrow[r]);       // arg <= 0 always
            }

            // ---- P: C-layout -> A-layout via LDS (wave-local DS ops are in-order)
#pragma unroll
            for (int r = 0; r < 8; ++r)
                pbuf[(r + half * 8) * 16 + ln16] = p[r];

            // ---- O += P @ V ; Osum += P @ ones  (K = 16 keys -> 4 chunks of 4)
            const float* vbase =
                levels + ((size_t)(b * NTOK + ky * GRID_W + kx0) * NLVL + l) * DIM;
#pragma unroll
            for (int kc = 0; kc < 4; ++kc) {
                int j0 = kc * 4 + half * 2;
                v2f pa = *(const v2f*)&pbuf[ln16 * 16 + j0];     // A-frag of P
                Osum = __builtin_amdgcn_wmma_f32_16x16x4_f32(
                           false, pa, false, onesb, (short)0, Osum, false, false);
#pragma unroll
                for (int dc = 0; dc < 8; ++dc) {
                    int d = dc * 16 + ln16;
                    v2f bv;                                      // B-frag of V: rows j0, j0+1
                    bv.x = vbase[(size_t)j0 * NLVL * DIM + d];
                    bv.y = vbase[(size_t)(j0 + 1) * NLVL * DIM + d];
                    O[dc] = __builtin_amdgcn_wmma_f32_16x16x4_f32(
                                false, pa, false, bv, (short)0, O[dc], false, false);
                }
            }
        }
    }

    // ---- epilogue: divide by softmax denom (row sums sit in Osum with matching layout)
    const int otok0 = qy * GRID_W + qx0;
#pragma unroll
    for (int r = 0; r < 8; ++r) {
        int   M   = r + half * 8;
        float inv = 1.0f / Osum[r];
        float* orow = out + ((size_t)(b * NTOK + otok0 + M) * NLVL + l) * DIM + ln16;
#pragma unroll
        for (int dc = 0; dc < 8; ++dc)
            orow[dc * 16] = O[dc][r] * inv;
    }
}

extern "C" void kernel_launch(void* const* d_in, const int* in_sizes, int n_in,
                              void* d_out, int out_size, void* d_ws, size_t ws_size,
                              hipStream_t stream) {
    (void)in_sizes; (void)n_in; (void)out_size; (void)d_ws; (void)ws_size;
    const float* levels = (const float*)d_in[0];
    float*       out    = (float*)d_out;
    // 64 query tiles per plane, 6 levels ("heads"), 8 batches; 1 wave32 per tile.
    dim3 grid(64, NLVL, BATCH);
    consensus_attn_kernel<<<grid, dim3(32, 1, 1), 0, stream>>>(levels, out);
}